// DownBlock_62947040690361
// MI455X (gfx1250) — compile-verified
//
#include <hip/hip_runtime.h>
#include <math.h>

// Problem sizes (fixed by the reference)
#define NNODE 50000
#define NEDGE 300000
#define NCN   12500
#define NCE   75000
#define WF    128
#define DEPTH 4

// A-operand construction modes
enum { A_PLAIN = 0, A_EDGE = 1, A_NODE = 2, A_POOL = 3, A_SCALAR = 4 };
// Epilogue modes
enum { E_SELU = 0, E_RES = 1, E_PLAIN = 2, E_SCAT = 3 };

typedef __attribute__((ext_vector_type(2))) float v2f;
typedef __attribute__((ext_vector_type(8))) float v8f;

// B is staged in LDS in fragment-pair layout: element (kr, n) of the [K,128]
// weight matrix lives at lds[(kr>>1)*PRS + n*2 + (kr&1)].  A WMMA B-fragment
// (B[kk][n], B[kk+1][n]) is then one aligned ds_load_b64.  PRS=288 floats:
// half-wave 1 reads pair-row +1 => byte offset 1152 => bank offset 32, so the
// two half-waves hit disjoint 32-bank groups (conflict-free b64 loads).
#define PRS 288

__device__ __forceinline__ float selu_f(float x) {
  const float scale = 1.0507009873554805f;
  const float alpha = 1.6732632423543772f;
  return scale * (x > 0.f ? x : alpha * (expf(x) - 1.f));
}

__device__ __forceinline__ v8f wmma4(v2f a, v2f b, v8f c) {
  return __builtin_amdgcn_wmma_f32_16x16x4_f32(false, a, false, b, (short)0, c,
                                               false, false);
}

// 128-column K-segment starting at absolute column kbeg (even).  Processed as
// 4 chunks of 8 k-steps with double-buffered A prefetch: the 8 global b64
// loads for chunk i+1 are issued before the 64 WMMAs of chunk i.
template <bool SC>
__device__ __forceinline__ void k_seg128(const float* __restrict__ base, float scale,
                                         int kbeg, const float* __restrict__ ldsB,
                                         int half, int l15, v8f* acc) {
  v2f abuf[2][8];
#pragma unroll
  for (int t = 0; t < 8; ++t)
    abuf[0][t] = *(const v2f*)(base + 4 * t + 2 * half);
#pragma unroll
  for (int ch = 0; ch < 4; ++ch) {
    if (ch < 3) {
#pragma unroll
      for (int t = 0; t < 8; ++t)
        abuf[(ch + 1) & 1][t] =
            *(const v2f*)(base + (ch + 1) * 32 + 4 * t + 2 * half);
    }
#pragma unroll
    for (int t = 0; t < 8; ++t) {
      v2f a = abuf[ch & 1][t];
      if (SC) { a.x *= scale; a.y *= scale; }
      const int r0 = (((kbeg + ch * 32 + 4 * t) >> 1) + half) * PRS + l15 * 2;
#pragma unroll
      for (int j = 0; j < 8; ++j) {
        v2f b = *(const v2f*)(ldsB + r0 + j * 32);
        acc[j] = wmma4(a, b, acc[j]);
      }
    }
  }
}

// One 4-wide K-step where the A columns are [sval, 0, 0, 0] (scalar feature
// column + zero padding).  LDS pair-rows beyond K are zero-filled.
__device__ __forceinline__ void k_step_scalar(float sval, int kbeg,
                                              const float* __restrict__ ldsB,
                                              int half, int l15, v8f* acc) {
  v2f a;
  a.x = (half == 0) ? sval : 0.f;
  a.y = 0.f;
  const int r0 = ((kbeg >> 1) + half) * PRS + l15 * 2;
#pragma unroll
  for (int j = 0; j < 8; ++j) {
    v2f b = *(const v2f*)(ldsB + r0 + j * 32);
    acc[j] = wmma4(a, b, acc[j]);
  }
}

// One wave computes one 16-row x 128-col stripe of OUT = epilogue(A @ B + bias).
// Exactly one tile per wave (no grid-stride loop => nothing for LICM to hoist,
// B fragments are consumed directly from LDS).  fp32 WMMA 16x16x4.
template <int AMODE, int EMODE>
__global__ __launch_bounds__(256) void gemm_mlp(
    const float* __restrict__ a0, const float* __restrict__ a1,
    const int* __restrict__ idx0, const int* __restrict__ idx1,
    const float* __restrict__ scal, const float* __restrict__ cntp,
    const float* __restrict__ B, const float* __restrict__ bias,
    const float* __restrict__ res, const int* __restrict__ sidx,
    float* __restrict__ out, int M, int K, int Kpad) {
  extern __shared__ float ldsB[];
  // cooperative weight staging into pair layout
  const int total = Kpad * WF;
  for (int i = threadIdx.x; i < total; i += blockDim.x) {
    int kr = i >> 7, c = i & (WF - 1);
    ldsB[(kr >> 1) * PRS + c * 2 + (kr & 1)] =
        (kr < K) ? B[(size_t)kr * WF + c] : 0.f;
  }
  __syncthreads();

  const int lane = threadIdx.x & 31;
  const int wave = threadIdx.x >> 5;
  const int half = lane >> 4;     // 0: lanes 0-15, 1: lanes 16-31
  const int l15  = lane & 15;
  const int tiles = (M + 15) >> 4;
  const int tile  = blockIdx.x * 8 + wave;
  if (tile >= tiles) return;

  const int m0 = tile << 4;
  int rA = m0 + l15;                 // row this lane feeds A from
  if (rA >= M) rA = M - 1;           // clamp (stores are guarded)

  v8f acc[8];
#pragma unroll
  for (int j = 0; j < 8; ++j) {
    float bz = bias[j * 16 + l15];
#pragma unroll
    for (int i = 0; i < 8; ++i) acc[j][i] = bz;
  }

  // ---- K loop, segmented per A construction mode ----
  if (AMODE == A_PLAIN) {
    k_seg128<false>(a0 + (size_t)rA * WF, 1.f, 0, ldsB, half, l15, acc);
  } else if (AMODE == A_EDGE) {
    const float* be = a0 + (size_t)rA * WF;
    const float* bs = a1 + (size_t)idx0[rA] * WF;
    const float* bd = a1 + (size_t)idx1[rA] * WF;
    k_seg128<false>(be, 1.f, 0, ldsB, half, l15, acc);
    k_seg128<false>(bs, 1.f, WF, ldsB, half, l15, acc);
    k_seg128<false>(bd, 1.f, 2 * WF, ldsB, half, l15, acc);
  } else if (AMODE == A_NODE) {
    float inv = 1.f / fmaxf(cntp[rA], 1.f);
    k_seg128<false>(a0 + (size_t)rA * WF, 1.f, 0, ldsB, half, l15, acc);
    k_seg128<true>(a1 + (size_t)rA * WF, inv, WF, ldsB, half, l15, acc);
  } else if (AMODE == A_POOL) {
    k_seg128<false>(a0 + (size_t)rA * WF, 1.f, 0, ldsB, half, l15, acc);
    k_step_scalar(scal[rA], WF, ldsB, half, l15, acc);   // col 128 + pad
  } else { // A_SCALAR (K = 1, padded to 4)
    k_step_scalar(scal[rA], 0, ldsB, half, l15, acc);
  }

  // epilogue: lane holds rows {i + 8*half}, col = 16j + l15
#pragma unroll
  for (int j = 0; j < 8; ++j) {
    const int col = j * 16 + l15;
#pragma unroll
    for (int i = 0; i < 8; ++i) {
      const int r = m0 + i + 8 * half;
      if (r < M) {
        float val = acc[j][i];
        if (EMODE == E_SELU) {
          out[(size_t)r * WF + col] = selu_f(val);
        } else if (EMODE == E_RES) {
          out[(size_t)r * WF + col] = res[(size_t)r * WF + col] + val;
        } else if (EMODE == E_PLAIN) {
          out[(size_t)r * WF + col] = val;
        } else { // E_SCAT: segment-sum into coarse nodes
          atomicAdd(&out[(size_t)sidx[r] * WF + col], val);
        }
      }
    }
  }
}

// ---------------- aux kernels ----------------

__global__ void count_kernel(const int* __restrict__ idx, int n,
                             float* __restrict__ cnt) {
  int i = blockIdx.x * blockDim.x + threadIdx.x;
  if (i < n) atomicAdd(&cnt[idx[i]], 1.0f);
}

__global__ void scatter_feat_kernel(const float* __restrict__ feat,
                                    const int* __restrict__ idx, int n,
                                    float* __restrict__ sum) {
  int i = blockIdx.x * blockDim.x + threadIdx.x;
  if (i < n * WF) {
    int r = i >> 7, c = i & (WF - 1);
    atomicAdd(&sum[(size_t)idx[r] * WF + c], feat[(size_t)r * WF + c]);
  }
}

__global__ void cpos_kernel(const float* __restrict__ pos,
                            const int* __restrict__ cluster, int n,
                            float* __restrict__ cposs, float* __restrict__ ccnt) {
  int i = blockIdx.x * blockDim.x + threadIdx.x;
  if (i < n) {
    int c = cluster[i];
    atomicAdd(&cposs[2 * c + 0], pos[2 * i + 0]);
    atomicAdd(&cposs[2 * c + 1], pos[2 * i + 1]);
    atomicAdd(&ccnt[c], 1.0f);
  }
}

__global__ void relfeat_kernel(const float* __restrict__ pos,
                               const int* __restrict__ cluster,
                               const float* __restrict__ cposs,
                               const float* __restrict__ ccnt, int n,
                               float* __restrict__ rf) {
  int i = blockIdx.x * blockDim.x + threadIdx.x;
  if (i < n) {
    int c = cluster[i];
    float inv = 1.f / fmaxf(ccnt[c], 1.f);
    float dx = pos[2 * i + 0] - cposs[2 * c + 0] * inv;
    float dy = pos[2 * i + 1] - cposs[2 * c + 1] * inv;
    rf[i] = sqrtf(dx * dx + dy * dy + 1e-12f);
  }
}

__global__ void crel_kernel(const int* __restrict__ cei,
                            const float* __restrict__ cposs,
                            const float* __restrict__ ccnt, int ec,
                            float* __restrict__ crf) {
  int i = blockIdx.x * blockDim.x + threadIdx.x;
  if (i < ec) {
    int a = cei[i], b = cei[ec + i];
    float ia = 1.f / fmaxf(ccnt[a], 1.f);
    float ib = 1.f / fmaxf(ccnt[b], 1.f);
    float dx = cposs[2 * a + 0] * ia - cposs[2 * b + 0] * ib;
    float dy = cposs[2 * a + 1] * ia - cposs[2 * b + 1] * ib;
    crf[i] = sqrtf(dx * dx + dy * dy + 1e-12f);
  }
}

__global__ void divide_kernel(float* __restrict__ data,
                              const float* __restrict__ cnt, int rows) {
  int i = blockIdx.x * blockDim.x + threadIdx.x;
  if (i < rows * WF) data[i] *= 1.f / fmaxf(cnt[i >> 7], 1.f);
}

// ---------------- host side ----------------

template <int AM, int EM>
static void run_gemm(const float* a0, const float* a1, const int* i0,
                     const int* i1, const float* sc, const float* cn,
                     const float* B, const float* bias, const float* res,
                     const int* sidx, float* out, int M, int K, int Kpad,
                     hipStream_t stream) {
  int tiles = (M + 15) / 16;
  int blocks = (tiles + 7) / 8;           // one tile per wave
  hipLaunchKernelGGL((gemm_mlp<AM, EM>), dim3(blocks), dim3(256),
                     (size_t)(Kpad / 2) * PRS * sizeof(float), stream,
                     a0, a1, i0, i1, sc, cn, B, bias, res, sidx, out, M, K, Kpad);
}

extern "C" void kernel_launch(void* const* d_in, const int* in_sizes, int n_in,
                              void* d_out, int out_size, void* d_ws, size_t ws_size,
                              hipStream_t stream) {
  (void)in_sizes; (void)n_in; (void)out_size; (void)ws_size;
  const float* v_in = (const float*)d_in[0];
  const float* e_in = (const float*)d_in[1];
  const float* pos  = (const float*)d_in[2];
  const int*   ei   = (const int*)d_in[3];
  const int*   clus = (const int*)d_in[4];
  const int*   cei  = (const int*)d_in[5];
  const float* ew0 = (const float*)d_in[6];   const float* eb0 = (const float*)d_in[7];
  const float* ew1 = (const float*)d_in[8];   const float* eb1 = (const float*)d_in[9];
  const float* ew2 = (const float*)d_in[10];  const float* eb2 = (const float*)d_in[11];
  const float* nw0 = (const float*)d_in[12];  const float* nb0 = (const float*)d_in[13];
  const float* nw1 = (const float*)d_in[14];  const float* nb1 = (const float*)d_in[15];
  const float* nw2 = (const float*)d_in[16];  const float* nb2 = (const float*)d_in[17];
  const float* pw0 = (const float*)d_in[18];  const float* pb0 = (const float*)d_in[19];
  const float* pw1 = (const float*)d_in[20];  const float* pb1 = (const float*)d_in[21];
  const float* pw2 = (const float*)d_in[22];  const float* pb2 = (const float*)d_in[23];
  const float* qw0 = (const float*)d_in[24];  const float* qb0 = (const float*)d_in[25];
  const float* qw1 = (const float*)d_in[26];  const float* qb1 = (const float*)d_in[27];
  const float* qw2 = (const float*)d_in[28];  const float* qb2 = (const float*)d_in[29];

  float* out = (float*)d_out;
  float* out_vc    = out;
  float* out_ec    = out_vc + (size_t)NCN * WF;
  float* out_vskip = out_ec + (size_t)NCE * WF;
  float* out_eskip = out_vskip + (size_t)NNODE * WF;

  const size_t E128 = (size_t)NEDGE * WF;
  const size_t N128 = (size_t)NNODE * WF;
  float* ws   = (float*)d_ws;
  float* X    = ws;            // e_cur
  float* Y    = X + E128;      // h0 / agg / eenc h0
  float* Z    = Y + E128;      // h1 / node h0 / pool h0
  float* vcur = Z + E128;
  float* cnt  = vcur + N128;   // N
  float* rf   = cnt + NNODE;   // N
  float* crf  = rf + NNODE;    // EC
  float* cposs = crf + NCE;    // NC*2
  float* ccnt  = cposs + (size_t)NCN * 2; // NC
  float* Zb = Z + N128;        // node/pool h1 (fits in Z's E*W span)
  float* Yb = Y + (size_t)NCE * WF; // eenc h1 (fits in Y's E*W span)

  const int* src = ei;
  const int* dst = ei + NEDGE;

  // dst degrees (constant across blocks)
  hipMemsetAsync(cnt, 0, NNODE * sizeof(float), stream);
  count_kernel<<<(NEDGE + 255) / 256, 256, 0, stream>>>(dst, NEDGE, cnt);

  for (int d = 0; d < DEPTH; ++d) {
    const float* ep = (d == 0) ? e_in : X;
    const float* vp = (d == 0) ? v_in : vcur;
    float* eo = (d == DEPTH - 1) ? out_eskip : X;
    float* vo = (d == DEPTH - 1) ? out_vskip : vcur;

    // edge MLP: e += mlp([e, v[src], v[dst]])
    run_gemm<A_EDGE, E_SELU>(ep, vp, src, dst, nullptr, nullptr,
        ew0 + (size_t)d * 3 * WF * WF, eb0 + d * WF, nullptr, nullptr,
        Y, NEDGE, 3 * WF, 3 * WF, stream);
    run_gemm<A_PLAIN, E_SELU>(Y, nullptr, nullptr, nullptr, nullptr, nullptr,
        ew1 + (size_t)d * WF * WF, eb1 + d * WF, nullptr, nullptr,
        Z, NEDGE, WF, WF, stream);
    run_gemm<A_PLAIN, E_RES>(Z, nullptr, nullptr, nullptr, nullptr, nullptr,
        ew2 + (size_t)d * WF * WF, eb2 + d * WF, ep, nullptr,
        eo, NEDGE, WF, WF, stream);

    // agg = seg_sum(e, dst)  (mean folded into node GEMM via cnt)
    hipMemsetAsync(Y, 0, N128 * sizeof(float), stream);
    scatter_feat_kernel<<<(int)((E128 + 255) / 256), 256, 0, stream>>>(eo, dst, NEDGE, Y);

    // node MLP: v += mlp([v, agg/deg])
    run_gemm<A_NODE, E_SELU>(vp, Y, nullptr, nullptr, nullptr, cnt,
        nw0 + (size_t)d * 2 * WF * WF, nb0 + d * WF, nullptr, nullptr,
        Z, NNODE, 2 * WF, 2 * WF, stream);
    run_gemm<A_PLAIN, E_SELU>(Z, nullptr, nullptr, nullptr, nullptr, nullptr,
        nw1 + (size_t)d * WF * WF, nb1 + d * WF, nullptr, nullptr,
        Zb, NNODE, WF, WF, stream);
    run_gemm<A_PLAIN, E_RES>(Zb, nullptr, nullptr, nullptr, nullptr, nullptr,
        nw2 + (size_t)d * WF * WF, nb2 + d * WF, vp, nullptr,
        vo, NNODE, WF, WF, stream);
  }

  // ---- MeshDownMP pooling ----
  hipMemsetAsync(cposs, 0, (size_t)NCN * 2 * sizeof(float), stream);
  hipMemsetAsync(ccnt, 0, NCN * sizeof(float), stream);
  cpos_kernel<<<(NNODE + 255) / 256, 256, 0, stream>>>(pos, clus, NNODE, cposs, ccnt);
  relfeat_kernel<<<(NNODE + 255) / 256, 256, 0, stream>>>(pos, clus, cposs, ccnt, NNODE, rf);

  // msg = mlp([v, relfeat])  (K = 129, padded to 132)
  run_gemm<A_POOL, E_SELU>(out_vskip, nullptr, nullptr, nullptr, rf, nullptr,
      pw0, pb0, nullptr, nullptr, Z, NNODE, WF + 1, WF + 4, stream);
  run_gemm<A_PLAIN, E_SELU>(Z, nullptr, nullptr, nullptr, nullptr, nullptr,
      pw1, pb1, nullptr, nullptr, Zb, NNODE, WF, WF, stream);
  hipMemsetAsync(out_vc, 0, (size_t)NCN * WF * sizeof(float), stream);
  run_gemm<A_PLAIN, E_SCAT>(Zb, nullptr, nullptr, nullptr, nullptr, nullptr,
      pw2, pb2, nullptr, clus, out_vc, NNODE, WF, WF, stream);
  divide_kernel<<<(NCN * WF + 255) / 256, 256, 0, stream>>>(out_vc, ccnt, NCN);

  // e_c = mlp(|coarse rel pos|)  (K = 1, padded to 4)
  crel_kernel<<<(NCE + 255) / 256, 256, 0, stream>>>(cei, cposs, ccnt, NCE, crf);
  run_gemm<A_SCALAR, E_SELU>(nullptr, nullptr, nullptr, nullptr, crf, nullptr,
      qw0, qb0, nullptr, nullptr, Y, NCE, 1, 4, stream);
  run_gemm<A_PLAIN, E_SELU>(Y, nullptr, nullptr, nullptr, nullptr, nullptr,
      qw1, qb1, nullptr, nullptr, Yb, NCE, WF, WF, stream);
  run_gemm<A_PLAIN, E_PLAIN>(Yb, nullptr, nullptr, nullptr, nullptr, nullptr,
      qw2, qb2, nullptr, nullptr, out_ec, NCE, WF, WF, stream);
}